// NCALoss_45569603010926
// MI455X (gfx1250) — compile-verified
//
#include <hip/hip_runtime.h>
#include <math.h>

#define NN 4096
#define DD 128
#define KSEL 32
#define ALPHA_ 16.0f

typedef __attribute__((ext_vector_type(2))) float v2f;
typedef __attribute__((ext_vector_type(8))) float v8f;

// ---------------------------------------------------------------------------
// Kernel 1: sim = X @ X^T  via V_WMMA_F32_16X16X4_F32 (full fp32 matrix op).
// Grid (32, 32), 256 threads (8 waves). Wave w computes a 16x128 strip:
// rows = blockIdx.y*128 + w*16 .. +15, cols = blockIdx.x*128 .. +127.
// A fragment (16x4 f32, 2 VGPRs): lane L -> m = L%16, k = ks + (L>>4)*2 + v
// B fragment (4x16 f32, 2 VGPRs): lane L -> n = L%16, same k mapping
// D (16x16 f32, 8 VGPRs):         lane L -> row = (L>>4)*8 + v, col = L%16
// ---------------------------------------------------------------------------
__global__ __launch_bounds__(256) void sim_gemm(const float* __restrict__ X,
                                                float* __restrict__ sim) {
  const int tid   = threadIdx.x;
  const int wave  = tid >> 5;
  const int lane  = tid & 31;
  const int lm    = lane & 15;
  const int hi    = lane >> 4;
  const int khalf = hi << 1;
  const int col0  = blockIdx.x * 128;
  const int row0  = blockIdx.y * 128 + wave * 16;

  v8f acc[8] = {};
  const float* arow = X + (size_t)(row0 + lm) * DD;

  for (int ks = 0; ks < DD; ks += 4) {
    const int k = ks + khalf;
    v2f a;
    a.x = arow[k];
    a.y = arow[k + 1];
#pragma unroll
    for (int s = 0; s < 8; ++s) {
      const float* brow = X + (size_t)(col0 + s * 16 + lm) * DD;
      v2f b;
      b.x = brow[k];
      b.y = brow[k + 1];
      acc[s] = __builtin_amdgcn_wmma_f32_16x16x4_f32(
          /*neg_a=*/false, a, /*neg_b=*/false, b,
          /*c_mod=*/(short)0, acc[s], /*reuse_a=*/false, /*reuse_b=*/false);
    }
  }

#pragma unroll
  for (int s = 0; s < 8; ++s) {
#pragma unroll
    for (int v = 0; v < 8; ++v) {
      const int gr = row0 + hi * 8 + v;
      const int gc = col0 + s * 16 + lm;
      sim[(size_t)gr * NN + gc] = acc[s][v];
    }
  }
}

// ---------------------------------------------------------------------------
// Block-wide reductions over 256 threads.
// ---------------------------------------------------------------------------
__device__ __forceinline__ float block_sum(float v, float* red, int tid) {
  __syncthreads();
  red[tid] = v;
  __syncthreads();
  for (int st = 128; st > 0; st >>= 1) {
    if (tid < st) red[tid] += red[tid + st];
    __syncthreads();
  }
  return red[0];
}

__device__ __forceinline__ float block_min(float v, float* red, int tid) {
  __syncthreads();
  red[tid] = v;
  __syncthreads();
  for (int st = 128; st > 0; st >>= 1) {
    if (tid < st) red[tid] = fminf(red[tid], red[tid + st]);
    __syncthreads();
  }
  return red[0];
}

// ---------------------------------------------------------------------------
// Kernel 2: per-row stats. One 256-thread block per row.
//  - row mean (base)
//  - exact K-th order statistic of "included" sims via 4-pass radix select
//  - pos/neg exp-weighted sums below threshold, min-pos fallback
//  - per-row loss term; last row also emits the diagnostics.
// ---------------------------------------------------------------------------
__global__ __launch_bounds__(256) void row_stats(const float* __restrict__ sim,
                                                 const int* __restrict__ tgt,
                                                 float* __restrict__ row_loss,
                                                 float* __restrict__ out4) {
  const int i   = blockIdx.x;
  const int tid = threadIdx.x;

  __shared__ float    srow[NN];          // 16 KB: the sim row
  __shared__ unsigned ukey[NN];          // 16 KB: monotone-mapped keys
  __shared__ unsigned char ssame[NN];    //  4 KB: same-class flags
  __shared__ float    red[256];
  __shared__ unsigned hist[256];
  __shared__ unsigned s_sel, s_want;

  const int ti = tgt[i];
  const float INF = __int_as_float(0x7F800000);

  // Load row, build selection keys, accumulate mean.
  float psum = 0.f;
  for (int t = 0; t < NN / 256; ++t) {
    const int j = tid + t * 256;
    const float s = sim[(size_t)i * NN + j];
    srow[j] = s;
    const bool same = (tgt[j] == ti);
    ssame[j] = same ? 1 : 0;
    // included = valid_pos | valid_neg = !(same && s >= 1)
    const bool inc = (!same) || (s < 1.0f);
    const unsigned u = __float_as_uint(inc ? s : INF);
    // monotone map: float order -> unsigned order
    ukey[j] = (u & 0x80000000u) ? ~u : (u | 0x80000000u);
    psum += s;
  }
  const float base = block_sum(psum, red, tid) * (1.0f / (float)NN);

  // Exact radix select of the (KSEL)-th smallest key (0-based), MSB first.
  unsigned prefix = 0u;
  unsigned want   = KSEL;
  for (int shift = 24; shift >= 0; shift -= 8) {
    hist[tid] = 0u;
    __syncthreads();
    const unsigned mask = (shift == 24) ? 0u : (0xFFFFFFFFu << (shift + 8));
    for (int t = 0; t < NN / 256; ++t) {
      const unsigned key = ukey[tid + t * 256];
      if ((key & mask) == prefix) atomicAdd(&hist[(key >> shift) & 255u], 1u);
    }
    __syncthreads();
    if (tid == 0) {
      unsigned cum = 0u, b = 0u;
      for (; b < 256u; ++b) {
        const unsigned h = hist[b];
        if (cum + h > want) break;
        cum += h;
      }
      s_sel  = b;
      s_want = want - cum;
    }
    __syncthreads();
    prefix |= (s_sel << shift);
    want    = s_want;
    __syncthreads();
  }
  const unsigned ub = (prefix & 0x80000000u) ? (prefix & 0x7FFFFFFFu) : ~prefix;
  const float thr = __uint_as_float(ub);   // exact K-th element value

  // Weighted sums below threshold + fallback min-pos.
  float pos_sum = 0.f, neg_sum = 0.f, poscnt = 0.f, minpos = INF;
  for (int t = 0; t < NN / 256; ++t) {
    const int j = tid + t * 256;
    const float s = srow[j];
    const bool same  = (ssame[j] != 0);
    const bool pos   = same && (s < 1.0f);
    const bool below = s < thr;
    const float w = expf(ALPHA_ * (base - s));
    if (pos && below)   { pos_sum += w; poscnt += 1.f; }
    if (!same && below) { neg_sum += w; }
    if (pos && s < minpos) minpos = s;
  }
  const float Ppos = block_sum(pos_sum, red, tid);
  const float Pneg = block_sum(neg_sum, red, tid);
  const float Pcnt = block_sum(poscnt, red, tid);
  const float Pmin = block_min(minpos, red, tid);

  const float pos_logit = (Pcnt > 0.f) ? Ppos : expf(ALPHA_ * (base - Pmin));
  const float lossi = -logf(pos_logit / (pos_logit + Pneg));
  if (tid == 0) row_loss[i] = lossi;

  // Diagnostics from the last row.
  if (i == NN - 1) {
    float ps = 0.f, pc = 0.f, ns = 0.f, nc = 0.f;
    for (int t = 0; t < NN / 256; ++t) {
      const int j = tid + t * 256;
      const float s = srow[j];
      const bool same = (ssame[j] != 0);
      if (same && (s < 1.0f)) { ps += s; pc += 1.f; }
      if (!same)              { ns += s; nc += 1.f; }
    }
    const float Sps = block_sum(ps, red, tid);
    const float Spc = block_sum(pc, red, tid);
    const float Sns = block_sum(ns, red, tid);
    const float Snc = block_sum(nc, red, tid);
    if (tid == 0) {
      out4[1] = 0.0f;          // prec (torch: c stays 0)
      out4[2] = Sps / Spc;     // mean_pos_sim
      out4[3] = Sns / Snc;     // mean_neg_sim
    }
  }
}

// ---------------------------------------------------------------------------
// Kernel 3: mean of per-row losses.
// ---------------------------------------------------------------------------
__global__ __launch_bounds__(256) void loss_reduce(const float* __restrict__ row_loss,
                                                   float* __restrict__ out4) {
  __shared__ float red[256];
  const int tid = threadIdx.x;
  float p = 0.f;
  for (int t = 0; t < NN / 256; ++t) p += row_loss[tid + t * 256];
  red[tid] = p;
  __syncthreads();
  for (int st = 128; st > 0; st >>= 1) {
    if (tid < st) red[tid] += red[tid + st];
    __syncthreads();
  }
  if (tid == 0) out4[0] = red[0] * (1.0f / (float)NN);
}

// ---------------------------------------------------------------------------
extern "C" void kernel_launch(void* const* d_in, const int* in_sizes, int n_in,
                              void* d_out, int out_size, void* d_ws, size_t ws_size,
                              hipStream_t stream) {
  const float* X   = (const float*)d_in[0];   // [4096,128] f32, L2-normalized
  const int*   tgt = (const int*)d_in[1];     // [4096] int32
  float* out = (float*)d_out;                 // 4 floats

  float* sim      = (float*)d_ws;                                   // 64 MB
  float* row_loss = (float*)((char*)d_ws + (size_t)NN * NN * sizeof(float));

  dim3 g1(NN / 128, NN / 128);
  sim_gemm<<<g1, 256, 0, stream>>>(X, sim);
  row_stats<<<NN, 256, 0, stream>>>(sim, tgt, row_loss, out);
  loss_reduce<<<1, 256, 0, stream>>>(row_loss, out);
}